// IR_CMPC_46918222741942
// MI455X (gfx1250) — compile-verified
//
#include <hip/hip_runtime.h>
#include <math.h>

#define BB 4096
#define DD 512
#define KK 4096
#define NN 131072
#define RR 3
#define GPAD 4160   // 4096 densities + pad slot for k+1 access

typedef __attribute__((ext_vector_type(2))) float v2f;
typedef __attribute__((ext_vector_type(8))) float v8f;

__device__ inline float block_reduce_sum(float v, float* sdata) {
  const int tid = threadIdx.x;
  sdata[tid] = v;
  __syncthreads();
  for (int s = blockDim.x >> 1; s > 0; s >>= 1) {
    if (tid < s) sdata[tid] += sdata[tid + s];
    __syncthreads();
  }
  float r = sdata[0];
  __syncthreads();
  return r;
}

// ---------------------------------------------------------------------------
// Normalize audio/frame rows; also rho0[i] = dot(an_i, fn_i)
// ---------------------------------------------------------------------------
__global__ __launch_bounds__(128) void norm_embed_kernel(
    const float* __restrict__ audio, const float* __restrict__ frame,
    float* __restrict__ an, float* __restrict__ fn, float* __restrict__ rho0) {
  __shared__ float sdata[128];
  const int i = blockIdx.x;
  const float* arow = audio + (size_t)i * DD;
  const float* frow = frame + (size_t)i * DD;
  float ssa = 0.f, ssf = 0.f, dot = 0.f;
  float av[4], fv[4];
#pragma unroll
  for (int q = 0; q < 4; ++q) {
    const int d = threadIdx.x + q * 128;
    av[q] = arow[d]; fv[q] = frow[d];
    ssa += av[q] * av[q]; ssf += fv[q] * fv[q]; dot += av[q] * fv[q];
  }
  ssa = block_reduce_sum(ssa, sdata);
  ssf = block_reduce_sum(ssf, sdata);
  dot = block_reduce_sum(dot, sdata);
  const float inva = 1.0f / fmaxf(sqrtf(ssa), 1e-12f);
  const float invf = 1.0f / fmaxf(sqrtf(ssf), 1e-12f);
#pragma unroll
  for (int q = 0; q < 4; ++q) {
    const int d = threadIdx.x + q * 128;
    an[(size_t)i * DD + d] = av[q] * inva;
    fn[(size_t)i * DD + d] = fv[q] * invf;
  }
  if (threadIdx.x == 0) rho0[i] = dot * inva * invf;
}

// ---------------------------------------------------------------------------
// Gather + normalize proto rows for 6 combos (dir 0: v2f uses f-clusters,
// dir 1: f2v uses a-clusters) and inverse-density arrays (padded).
// ---------------------------------------------------------------------------
__global__ __launch_bounds__(128) void gather_protos_kernel(
    const float* __restrict__ a_cent, const float* __restrict__ a_dens,
    const int* __restrict__ a_i2c,
    const float* __restrict__ f_cent, const float* __restrict__ f_dens,
    const int* __restrict__ f_i2c,
    const int* __restrict__ vindex,
    float* __restrict__ P_all, float* __restrict__ gin_all) {
  __shared__ float sdata[128];
  const int j = blockIdx.x;
  const int c = blockIdx.y;    // 0..5
  const int dir = c / 3;
  const int r = c % 3;
  const float* cent = (dir == 0) ? f_cent : a_cent;
  const float* dens = (dir == 0) ? f_dens : a_dens;
  const int* i2c   = (dir == 0) ? f_i2c : a_i2c;
  const int pid = i2c[(size_t)r * NN + vindex[j]];
  const float* src = cent + ((size_t)r * KK + pid) * DD;
  float ss = 0.f; float sv[4];
#pragma unroll
  for (int q = 0; q < 4; ++q) {
    const int d = threadIdx.x + q * 128;
    sv[q] = src[d]; ss += sv[q] * sv[q];
  }
  ss = block_reduce_sum(ss, sdata);
  const float inv = 1.0f / fmaxf(sqrtf(ss), 1e-12f);
  float* Pc = P_all + ((size_t)c * BB + j) * DD;
#pragma unroll
  for (int q = 0; q < 4; ++q) {
    const int d = threadIdx.x + q * 128;
    Pc[d] = sv[q] * inv;
  }
  if (threadIdx.x == 0) {
    gin_all[(size_t)c * GPAD + j] = 1.0f / dens[(size_t)r * KK + pid];
    if (j == 0) gin_all[(size_t)c * GPAD + BB] = 1.0f;  // pad for k+1 loads
  }
}

// ---------------------------------------------------------------------------
// rho[i] = rho0[i] - sum_r dot(a_cent[r][apid], f_cent[r][fpid])  (unnormalized)
// ---------------------------------------------------------------------------
__global__ __launch_bounds__(128) void rho_kernel(
    const float* __restrict__ a_cent, const int* __restrict__ a_i2c,
    const float* __restrict__ f_cent, const int* __restrict__ f_i2c,
    const int* __restrict__ vindex, const float* __restrict__ rho0,
    float* __restrict__ rho) {
  __shared__ float sdata[128];
  const int i = blockIdx.x;
  const int vid = vindex[i];
  float acc = 0.f;
  for (int r = 0; r < RR; ++r) {
    const int ap = a_i2c[(size_t)r * NN + vid];
    const int fp = f_i2c[(size_t)r * NN + vid];
    const float* ca = a_cent + ((size_t)r * KK + ap) * DD;
    const float* cf = f_cent + ((size_t)r * KK + fp) * DD;
#pragma unroll
    for (int q = 0; q < 4; ++q) {
      const int d = threadIdx.x + q * 128;
      acc += ca[d] * cf[d];
    }
  }
  acc = block_reduce_sum(acc, sdata);
  if (threadIdx.x == 0) rho[i] = rho0[i] - acc;
}

// ---------------------------------------------------------------------------
// Fused GEMM (fp32 WMMA 16x16x4) + per-row online logsumexp.
// Tiles staged into LDS with GLOBAL_LOAD_ASYNC_TO_LDS_B128 (double buffered,
// ASYNCcnt-synchronized); fragments consumed via ds_load_b64.
// Block: 8 waves = 128 rows (16/wave), shared 64-column tiles.
// Nested loops: acc reset + LSE epilogue outside the 16-stage K loop so the
// steady-state body is pure ds_load + wmma (no per-stage selects).
// ---------------------------------------------------------------------------
#define TSTR 36                 // padded LDS row stride (floats); 144B, 16B-aligned
#define ABUF (128 * TSTR)       // one A buffer (floats)
#define BBUF (64 * TSTR)        // one B buffer (floats)
#define BBASE (2 * ABUF)
#define NJT (BB / 64)           // 64 column tiles
#define NKS (DD / 32)           // 16 K stages per column tile
#define NST (NJT * NKS)         // 1024 total stages

__global__ __launch_bounds__(256) void gemm_lse_kernel(
    const float* __restrict__ A, const float* __restrict__ Bm,
    const float* __restrict__ ginv, float* __restrict__ lossOut, int mode) {
  __shared__ float lds[2 * ABUF + 2 * BBUF];

  const int t = threadIdx.x;
  const int wv = t >> 5;
  const int lane = t & 31;
  const int half = lane >> 4;
  const int l16 = lane & 15;
  const int i0b = blockIdx.x * 128;
  const int i0 = i0b + wv * 16;
  const float invT = 1.0f / 0.07f;
  const float g0inv = mode ? ginv[0] : invT;

  const uint32_t lds_base = (uint32_t)(uintptr_t)(&lds[0]);

  float mrow[8], srow[8], dacc[8];
#pragma unroll
  for (int v = 0; v < 8; ++v) { mrow[v] = -3.0e38f; srow[v] = 0.f; dacc[v] = 0.f; }

  const v8f vzero = {0.f, 0.f, 0.f, 0.f, 0.f, 0.f, 0.f, 0.f};

  // Issue one stage of async tile fills: A 128x32 (4 x B128/thread),
  // B 64x32 (2 x B128/thread). 6 async loads per thread per stage.
  auto issue = [&](int stage) {
    const int jt = (stage >> 4) * 64;
    const int kc = (stage & 15) * 32;
    const int b = stage & 1;
    const uint32_t aoff = lds_base + (uint32_t)(b * ABUF) * 4u;
    const uint32_t boff = lds_base + (uint32_t)(BBASE + b * BBUF) * 4u;
#pragma unroll
    for (int q = 0; q < 4; ++q) {
      const int c = t + q * 256;           // chunk id, 1024 total
      const int row = c >> 3;
      const int cb = (c & 7) * 4;          // float offset within the 32-float slab
      const float* g = A + (size_t)(i0b + row) * DD + kc + cb;
      const uint32_t l = aoff + (uint32_t)(row * TSTR + cb) * 4u;
      asm volatile("global_load_async_to_lds_b128 %0, %1, off"
                   :: "v"(l), "v"(g) : "memory");
    }
#pragma unroll
    for (int q = 0; q < 2; ++q) {
      const int c = t + q * 256;           // chunk id, 512 total
      const int row = c >> 3;
      const int cb = (c & 7) * 4;
      const float* g = Bm + (size_t)(jt + row) * DD + kc + cb;
      const uint32_t l = boff + (uint32_t)(row * TSTR + cb) * 4u;
      asm volatile("global_load_async_to_lds_b128 %0, %1, off"
                   :: "v"(l), "v"(g) : "memory");
    }
  };

  issue(0);
  for (int jti = 0; jti < NJT; ++jti) {
    v8f acc[4];
#pragma unroll
    for (int cc = 0; cc < 4; ++cc) acc[cc] = vzero;

    for (int ks = 0; ks < NKS; ++ks) {
      const int s = jti * NKS + ks;
      if (s + 1 < NST) {
        issue(s + 1);
        // 6 just-issued outstanding; previous stage's 6 retire in order.
        asm volatile("s_wait_asynccnt 0x6" ::: "memory");
      } else {
        asm volatile("s_wait_asynccnt 0x0" ::: "memory");
      }
      __syncthreads();   // all waves' stage-s tile data visible in LDS

      const int b = s & 1;
      // Fragment bases (conflict-free: 36-float stride -> 16 rows, 16 banks)
      const float* ab = &lds[b * ABUF] + (size_t)(wv * 16 + l16) * TSTR + 2 * half;
      const float* bb = &lds[BBASE + b * BBUF] + (size_t)l16 * TSTR + 2 * half;
#pragma unroll
      for (int k0 = 0; k0 < 32; k0 += 4) {
        v2f af = *(const v2f*)(ab + k0);
        v2f b0 = *(const v2f*)(bb + k0);
        v2f b1 = *(const v2f*)(bb + 16 * TSTR + k0);
        v2f b2 = *(const v2f*)(bb + 32 * TSTR + k0);
        v2f b3 = *(const v2f*)(bb + 48 * TSTR + k0);
        acc[0] = __builtin_amdgcn_wmma_f32_16x16x4_f32(false, af, false, b0, (short)0, acc[0], false, false);
        acc[1] = __builtin_amdgcn_wmma_f32_16x16x4_f32(false, af, false, b1, (short)0, acc[1], false, false);
        acc[2] = __builtin_amdgcn_wmma_f32_16x16x4_f32(false, af, false, b2, (short)0, acc[2], false, false);
        acc[3] = __builtin_amdgcn_wmma_f32_16x16x4_f32(false, af, false, b3, (short)0, acc[3], false, false);
      }
      __syncthreads();   // protect buffer reuse before next issue()
    }

    // Epilogue for the finished 64-column tile: scaled logits + online LSE.
    {
      const int jt = jti * 64;
#pragma unroll
      for (int cc = 0; cc < 4; ++cc) {
        const int k = jt + cc * 16 + l16;   // global column for this lane
        float gk = invT, gk1 = invT;
        if (mode) { gk = ginv[k]; gk1 = ginv[k + 1]; }
#pragma unroll
        for (int v = 0; v < 8; ++v) {
          const int irow = i0 + v + 8 * half;  // C layout: M=v / M=v+8 per half
          const float sf = mode ? ((k < irow) ? gk1 : ((k == irow) ? g0inv : gk)) : invT;
          const float logit = acc[cc][v] * sf;
          float lmax = logit;
#pragma unroll
          for (int msk = 1; msk <= 8; msk <<= 1)
            lmax = fmaxf(lmax, __shfl_xor(lmax, msk, 32));
          const float nm = fmaxf(mrow[v], lmax);
          float e = __expf(logit - nm);
          float dsel = (k == irow) ? logit : 0.0f;
#pragma unroll
          for (int msk = 1; msk <= 8; msk <<= 1) {
            e += __shfl_xor(e, msk, 32);
            dsel += __shfl_xor(dsel, msk, 32);
          }
          srow[v] = srow[v] * __expf(mrow[v] - nm) + e;
          mrow[v] = nm;
          dacc[v] += dsel;
        }
      }
    }
  }

  if (l16 == 0) {
#pragma unroll
    for (int v = 0; v < 8; ++v)
      lossOut[i0 + v + 8 * half] = logf(srow[v]) + mrow[v] - dacc[v];
  }
}

// ---------------------------------------------------------------------------
// Single-block: bitonic argsort of rho, mean/std, Gaussian pdf, cumsum,
// scatter weight back by rank.
// ---------------------------------------------------------------------------
__global__ __launch_bounds__(1024) void weights_kernel(
    const float* __restrict__ rho, float* __restrict__ wgt) {
  __shared__ float key[BB];
  __shared__ int idx[BB];
  __shared__ float red[1024];
  const int tid = threadIdx.x;
  float lsum = 0.f, lsq = 0.f;
#pragma unroll
  for (int q = 0; q < 4; ++q) {
    const int i = tid + q * 1024;
    const float x = rho[i];
    key[i] = x; idx[i] = i;
    lsum += x; lsq += x * x;
  }
  __syncthreads();
  red[tid] = lsum; __syncthreads();
  for (int s = 512; s > 0; s >>= 1) { if (tid < s) red[tid] += red[tid + s]; __syncthreads(); }
  const float mean = red[0] / (float)BB; __syncthreads();
  red[tid] = lsq; __syncthreads();
  for (int s = 512; s > 0; s >>= 1) { if (tid < s) red[tid] += red[tid + s]; __syncthreads(); }
  const float var = red[0] / (float)BB - mean * mean; __syncthreads();
  const float sd = sqrtf(fmaxf(var, 0.f));
  const float mu = mean + 0.3f * sd;        // DELTA
  const float sigma = sd * sqrtf(2.0f);     // std * sqrt(KA)

  // bitonic sort ascending (argsort)
  for (int k = 2; k <= BB; k <<= 1) {
    for (int j = k >> 1; j > 0; j >>= 1) {
      for (int q = 0; q < 4; ++q) {
        const int i = tid + q * 1024;
        const int ixj = i ^ j;
        if (ixj > i) {
          const bool up = ((i & k) == 0);
          const float a = key[i], b = key[ixj];
          if (up ? (a > b) : (a < b)) {
            key[i] = b; key[ixj] = a;
            const int t = idx[i]; idx[i] = idx[ixj]; idx[ixj] = t;
          }
        }
      }
      __syncthreads();
    }
  }

  // Gaussian pdf over sorted rho + inclusive cumsum (chunks of 4 per thread)
  const float inv_sigma = 1.0f / sigma;
  const float coef = 0.3989422804014327f * inv_sigma;  // 1/(sqrt(2pi)*sigma)
  float yv[4]; float lt = 0.f;
#pragma unroll
  for (int q = 0; q < 4; ++q) {
    const int i = tid * 4 + q;
    const float z = (key[i] - mu) * inv_sigma;
    lt += coef * __expf(-0.5f * z * z);
    yv[q] = lt;
  }
  __syncthreads();
  red[tid] = lt; __syncthreads();
  for (int off = 1; off < 1024; off <<= 1) {
    const float v = (tid >= off) ? red[tid - off] : 0.f;
    __syncthreads();
    red[tid] += v;
    __syncthreads();
  }
  const float excl = red[tid] - lt;
  const float total = red[1023];
#pragma unroll
  for (int q = 0; q < 4; ++q) {
    const int i = tid * 4 + q;
    wgt[idx[i]] = (excl + yv[q]) / total;  // w[order[t]] = y_cum[t]/y_cum[-1]
  }
}

// ---------------------------------------------------------------------------
// Final weighted sums -> scalar loss. ip weights from in-loop /3: 1/27,1/9,1/3.
// ---------------------------------------------------------------------------
__global__ __launch_bounds__(256) void finalize_kernel(
    const float* __restrict__ wgt,
    const float* __restrict__ Lv_ii, const float* __restrict__ Lf_ii,
    const float* __restrict__ Lip_all, float* __restrict__ out) {
  __shared__ float sdata[256];
  const float c0 = 1.0f / 27.0f, c1 = 1.0f / 9.0f, c2 = 1.0f / 3.0f;
  float sw = 0.f, svv = 0.f, sff = 0.f;
  for (int i = threadIdx.x; i < BB; i += 256) {
    const float w = wgt[i];
    const float lv = Lv_ii[i] + c0 * Lip_all[i] + c1 * Lip_all[BB + i] + c2 * Lip_all[2 * BB + i];
    const float lf = Lf_ii[i] + c0 * Lip_all[3 * BB + i] + c1 * Lip_all[4 * BB + i] + c2 * Lip_all[5 * BB + i];
    sw += w; svv += w * lv; sff += w * lf;
  }
  sw = block_reduce_sum(sw, sdata);
  svv = block_reduce_sum(svv, sdata);
  sff = block_reduce_sum(sff, sdata);
  if (threadIdx.x == 0) out[0] = svv / sw + sff / sw;
}

extern "C" void kernel_launch(void* const* d_in, const int* in_sizes, int n_in,
                              void* d_out, int out_size, void* d_ws, size_t ws_size,
                              hipStream_t stream) {
  const float* audio  = (const float*)d_in[0];
  const float* frame  = (const float*)d_in[1];
  const float* a_cent = (const float*)d_in[2];
  const float* a_dens = (const float*)d_in[3];
  const float* f_cent = (const float*)d_in[4];
  const float* f_dens = (const float*)d_in[5];
  const int* vindex   = (const int*)d_in[6];
  const int* a_i2c    = (const int*)d_in[7];
  const int* f_i2c    = (const int*)d_in[8];
  float* out = (float*)d_out;
  float* ws = (float*)d_ws;

  size_t off = 0;
  float* an      = ws + off; off += (size_t)BB * DD;
  float* fn      = ws + off; off += (size_t)BB * DD;
  float* P_all   = ws + off; off += (size_t)6 * BB * DD;
  float* gin_all = ws + off; off += (size_t)6 * GPAD;
  float* rho0    = ws + off; off += BB;
  float* rho     = ws + off; off += BB;
  float* wgt     = ws + off; off += BB;
  float* Lv_ii   = ws + off; off += BB;
  float* Lf_ii   = ws + off; off += BB;
  float* Lip_all = ws + off; off += (size_t)6 * BB;

  norm_embed_kernel<<<BB, 128, 0, stream>>>(audio, frame, an, fn, rho0);
  gather_protos_kernel<<<dim3(BB, 6), 128, 0, stream>>>(
      a_cent, a_dens, a_i2c, f_cent, f_dens, f_i2c, vindex, P_all, gin_all);
  rho_kernel<<<BB, 128, 0, stream>>>(a_cent, a_i2c, f_cent, f_i2c, vindex, rho0, rho);

  // inst-inst (both directions)
  gemm_lse_kernel<<<BB / 128, 256, 0, stream>>>(an, fn, nullptr, Lv_ii, 0);
  gemm_lse_kernel<<<BB / 128, 256, 0, stream>>>(fn, an, nullptr, Lf_ii, 0);
  // inst-proto: v2f (f-clusters) then f2v (a-clusters), r = 0..2
  for (int r = 0; r < 3; ++r) {
    gemm_lse_kernel<<<BB / 128, 256, 0, stream>>>(
        an, P_all + (size_t)r * BB * DD, gin_all + (size_t)r * GPAD,
        Lip_all + (size_t)r * BB, 1);
    gemm_lse_kernel<<<BB / 128, 256, 0, stream>>>(
        fn, P_all + (size_t)(3 + r) * BB * DD, gin_all + (size_t)(3 + r) * GPAD,
        Lip_all + (size_t)(3 + r) * BB, 1);
  }

  weights_kernel<<<1, 1024, 0, stream>>>(rho, wgt);
  finalize_kernel<<<1, 256, 0, stream>>>(wgt, Lv_ii, Lf_ii, Lip_all, out);
}